// DivMergedLayer1_46548855554094
// MI455X (gfx1250) — compile-verified
//
#include <hip/hip_runtime.h>

// ---------------------------------------------------------------------------
// MI455X (gfx1250) implementation.
//
// out = x + [relu(x Wup^T) * (x Wgate^T)] Wdown^T   (combined c/g FFN, H=41)
//       + scalar softmax correction at column SLOT_QUOTIENT=5 of row 0.
//
// f32 WMMA (V_WMMA_F32_16X16X4_F32): gate weights reach 2^31 (overflows f16),
// and the reference accumulates in f32. Kernel is memory-bound (256 MiB
// traffic ~ 11.5us @ 23.3 TB/s); the ~8 GF of matrix work rides the WMMA pipe.
// x is staged global->LDS with CDNA5 async-to-LDS copies (ASYNCcnt), read
// from HBM exactly once, and all WMMA operand fetches are bank-conflict-free.
// ---------------------------------------------------------------------------

typedef float v2f __attribute__((ext_vector_type(2)));
typedef float v4f __attribute__((ext_vector_type(4)));
typedef float v8f __attribute__((ext_vector_type(8)));

#define FDIM    4096      // N*D flattened feature dim
#define BT      16        // batch rows per workgroup
#define HPAD    64        // hidden width padded 41 -> 64 (3 live 16-wide tiles)
#define XSTRIDE 4100      // x tile row stride in floats (bank-conflict-free)
#define ASTRIDE 68        // act row stride in floats (bank-conflict-free)
// LDS layout (float offsets)
#define P_OFF   (BT * XSTRIDE)            // partials: 4 waves * 6 tiles * 256
#define A_OFF   (P_OFF + 4 * 6 * 256)     // act: 16 x ASTRIDE
#define C_OFF   (A_OFF + BT * ASTRIDE)    // corr: 16
#define LDS_FLOATS (C_OFF + BT)
#define LDS_BYTES  (LDS_FLOATS * 4)       // 291,392 B  (< 320 KB/WGP)

// --- pack the 6 sparse weight arrays into WMMA-friendly layouts -------------
// K/j-pair interleaved so each lane's two B-matrix values (VGPR0/VGPR1 = two
// adjacent K rows) are 8 contiguous bytes -> single global_load_b64 per WMMA.
//   wupP/wgateP : [k/2][j][2]  -> idx = (k>>1)*2*HPAD + 2*j + (k&1)
//   wdownP      : [j/2][k][2]  -> idx = (j>>1)*2*FDIM + 2*k + (j&1)
__global__ void pack_weights_kernel(const float* __restrict__ cWup,
                                    const float* __restrict__ cWgate,
                                    const float* __restrict__ cWdown,
                                    const float* __restrict__ gWup,
                                    const float* __restrict__ gWgate,
                                    const float* __restrict__ gWdown,
                                    float* __restrict__ wupP,
                                    float* __restrict__ wgateP,
                                    float* __restrict__ wdownP) {
  const int k = blockIdx.x;     // 0..4095
  const int j = threadIdx.x;    // 0..63
  float u = 0.f, g = 0.f, d = 0.f;
  if (j < 8) {
    u = cWup[j * FDIM + k];
    g = cWgate[j * FDIM + k];
    d = cWdown[k * 8 + j];
  } else if (j < 41) {
    u = gWup[(j - 8) * FDIM + k];
    g = gWgate[(j - 8) * FDIM + k];
    d = gWdown[k * 33 + (j - 8)];
  }
  wupP[(k >> 1) * (2 * HPAD) + 2 * j + (k & 1)]   = u;
  wgateP[(k >> 1) * (2 * HPAD) + 2 * j + (k & 1)] = g;
  wdownP[(j >> 1) * (2 * FDIM) + 2 * k + (j & 1)] = d;
}

// --- main fused kernel: one WG per 16 batch rows, 4 waves -------------------
__global__ void __launch_bounds__(128)
ffn_div_kernel(const float* __restrict__ x,
               const float* __restrict__ wupP,
               const float* __restrict__ wgateP,
               const float* __restrict__ wdownP,
               float* __restrict__ out) {
  extern __shared__ float lds[];
  const int tid  = threadIdx.x;
  const int wave = tid >> 5;
  const int lane = tid & 31;
  const int m    = lane & 15;   // low half-lane index (row for A, col for B/C)
  const int hi   = lane >> 4;   // 0 = lanes 0-15, 1 = lanes 16-31
  const long b0  = (long)blockIdx.x * BT;

  // ---- Phase 0: async global->LDS staging of the 16 x 4096 x-tile ----
  // CDNA5 GLOBAL_LOAD_ASYNC_TO_LDS_B128: no VGPR round-trip, ASYNCcnt-tracked.
  for (int idx = tid; idx < BT * (FDIM / 4); idx += 128) {
    const int row = idx >> 10;          // 1024 float4 per row
    const int c4  = idx & 1023;
    const float* gptr = x + (b0 + row) * FDIM + (long)c4 * 4;
    const unsigned ldsoff = (unsigned)((row * XSTRIDE + c4 * 4) * 4); // bytes, 16B aligned
    asm volatile("global_load_async_to_lds_b128 %0, %1, off"
                 :: "v"(ldsoff), "v"(gptr) : "memory");
  }
  asm volatile("s_wait_asynccnt 0x0" ::: "memory");
  __syncthreads();

  // ---- Phase 1: up/gate GEMM. Each wave covers a 1024-wide K slice. ----
  v8f accU[3], accG[3];
#pragma unroll
  for (int t = 0; t < 3; ++t) { accU[t] = (v8f)0.f; accG[t] = (v8f)0.f; }

  const int kbeg = wave * 1024;
  for (int kk = kbeg; kk < kbeg + 1024; kk += 4) {
    // A (16x4 of x): lane holds x[m][kk+2*hi .. +1], conflict-free ds_load_b64
    v2f a = *(const v2f*)(lds + m * XSTRIDE + kk + 2 * hi);
    // kk+2*hi is even -> paired layout gives both K rows in one b64 load
    const float* bu = wupP   + ((kk >> 1) + hi) * (2 * HPAD) + 2 * m;
    const float* bg = wgateP + ((kk >> 1) + hi) * (2 * HPAD) + 2 * m;
#pragma unroll
    for (int t = 0; t < 3; ++t) {
      v2f b = *(const v2f*)(bu + 32 * t);     // {W[K][j], W[K+1][j]}
      accU[t] = __builtin_amdgcn_wmma_f32_16x16x4_f32(
          false, a, false, b, (short)0, accU[t], false, false);
      b = *(const v2f*)(bg + 32 * t);
      accG[t] = __builtin_amdgcn_wmma_f32_16x16x4_f32(
          false, a, false, b, (short)0, accG[t], false, false);
    }
  }

  // store per-wave partial tiles (raw lane-major) for cross-wave reduction
#pragma unroll
  for (int t = 0; t < 3; ++t) {
    float* pu = lds + P_OFF + (wave * 6 + t) * 256 + lane * 8;
    float* pg = lds + P_OFF + (wave * 6 + 3 + t) * 256 + lane * 8;
#pragma unroll
    for (int v = 0; v < 8; ++v) { pu[v] = accU[t][v]; pg[v] = accG[t][v]; }
  }
  __syncthreads();

  // ---- Phase 1b: waves 0-2 reduce + activation; wave 3 does f64 softmax ----
  if (wave < 3) {
    const int t = wave;
    float u[8], g[8];
#pragma unroll
    for (int v = 0; v < 8; ++v) { u[v] = 0.f; g[v] = 0.f; }
    for (int w = 0; w < 4; ++w) {
      const float* pu = lds + P_OFF + (w * 6 + t) * 256 + lane * 8;
      const float* pg = lds + P_OFF + (w * 6 + 3 + t) * 256 + lane * 8;
#pragma unroll
      for (int v = 0; v < 8; ++v) { u[v] += pu[v]; g[v] += pg[v]; }
    }
    // act = relu(up) * gate, written in C-layout positions -> [row][j]
#pragma unroll
    for (int v = 0; v < 8; ++v) {
      const int row = v + 8 * hi;
      lds[A_OFF + row * ASTRIDE + t * 16 + m] = fmaxf(u[v], 0.f) * g[v];
    }
  } else if (lane < 16) {
    // exact f64 softmax-recip correction, mirrors the reference
    const int row = lane;
    double mx = -1.0e300;
    for (int i = 0; i < 32; ++i) {
      double dd = (double)lds[row * XSTRIDE + i * 128 + 1];     // NIB_B
      double s  = (dd > 0.5) ? log(dd * (double)(1u << i)) : -60.0;
      mx = fmax(mx, s);
    }
    double sum = 0.0;
    for (int i = 0; i < 32; ++i) {
      double dd = (double)lds[row * XSTRIDE + i * 128 + 1];
      double s  = (dd > 0.5) ? log(dd * (double)(1u << i)) : -60.0;
      sum += exp(s - mx);
    }
    const double recip = exp(-mx) / fmax(sum, 1e-30);
    // op_w = x[b,0,OP_START+OPCODE] = x[b, col 67]
    lds[C_OFF + row] = lds[row * XSTRIDE + 67] * (float)recip;
  }
  __syncthreads();

  // ---- Phase 2: down GEMM (16x48 act @ 48x4096), fused out = x + d12 ----
  v2f av[12];
#pragma unroll
  for (int s = 0; s < 12; ++s)
    av[s] = *(const v2f*)(lds + A_OFF + m * ASTRIDE + 4 * s + 2 * hi);

  for (int tile = wave; tile < 256; tile += 4) {
    const int nb = tile * 16;
    v8f c = (v8f)0.f;
#pragma unroll
    for (int s = 0; s < 12; ++s) {
      // j0 = 4s+2*hi is even -> one b64 gives {down[j0][n], down[j0+1][n]}
      v2f b = *(const v2f*)(wdownP + (2 * s + hi) * (2 * FDIM) + 2 * (nb + m));
      c = __builtin_amdgcn_wmma_f32_16x16x4_f32(
          false, av[s], false, b, (short)0, c, false, false);
    }
#pragma unroll
    for (int v = 0; v < 8; ++v) {
      const int row = v + 8 * hi;
      float val = c[v] + lds[row * XSTRIDE + nb + m];
      if (nb == 0 && m == 5) val += lds[C_OFF + row];   // SLOT_QUOTIENT col 5
      out[(b0 + row) * FDIM + nb + m] = val;
    }
  }
}

extern "C" void kernel_launch(void* const* d_in, const int* in_sizes, int n_in,
                              void* d_out, int out_size, void* d_ws, size_t ws_size,
                              hipStream_t stream) {
  (void)in_sizes; (void)n_in; (void)out_size; (void)ws_size;
  const float* x      = (const float*)d_in[0];
  const float* cWup   = (const float*)d_in[1];
  const float* cWgate = (const float*)d_in[2];
  const float* cWdown = (const float*)d_in[3];
  const float* gWup   = (const float*)d_in[4];
  const float* gWgate = (const float*)d_in[5];
  const float* gWdown = (const float*)d_in[6];
  // d_in[7] = base_powers (f64) -- recomputed exactly on device, unused here.
  float* out = (float*)d_out;

  // workspace: packed weights, 3 * 4096 * 64 floats = 3 MiB
  float* wupP   = (float*)d_ws;
  float* wgateP = wupP + FDIM * HPAD;
  float* wdownP = wgateP + FDIM * HPAD;

  pack_weights_kernel<<<FDIM, HPAD, 0, stream>>>(
      cWup, cWgate, cWdown, gWup, gWgate, gWdown, wupP, wgateP, wdownP);

  // allow > 64 KB dynamic LDS (idempotent, capture-safe host-side attribute)
  hipFuncSetAttribute((const void*)ffn_div_kernel,
                      hipFuncAttributeMaxDynamicSharedMemorySize, LDS_BYTES);

  ffn_div_kernel<<<8192 / BT, 128, LDS_BYTES, stream>>>(
      x, wupP, wgateP, wdownP, out);
}